// ContrastiveLoss_52561809768711
// MI455X (gfx1250) — compile-verified
//
#include <hip/hip_runtime.h>
#include <hip/hip_bf16.h>

// CDNA5 / gfx1250, wave32. V_WMMA_F32_16X16X4_F32 (fp32 matrix pipe) keeps
// exact fp32 numerics (output goes through exp(x/0.07), so bf16 downcast
// would be numerically unfaithful).
//
// Blocking: each workgroup handles 4 text batches (b0..b0+3) x 1 img batch d.
//  - img tile staged once in LDS feeds 4 output rows  -> 4x less L2 traffic
//  - B fragment reused by 4 independent WMMA/accs     -> fewer DS loads/wmma,
//    no RAW chain between consecutive v_wmma instructions.

typedef __attribute__((ext_vector_type(2))) float v2f;
typedef __attribute__((ext_vector_type(8))) float v8f;

#define NB   128          // batch (b and d)
#define NP   100          // proposals
#define NT   16           // phrases
#define NF   512          // feature dim
#define NBB  4            // text batches per workgroup
#define KB   64           // K tile
#define LDA  (KB + 4)     // LDS row stride (floats); +4 pad -> bank-conflict free
#define LDB  (KB + 4)
#define NPP  112          // proposals padded to 7*16

__global__ __launch_bounds__(256)
void ContrastiveLoss_52561809768711_kernel(const float* __restrict__ img,
                                           const float* __restrict__ text,
                                           float* __restrict__ out) {
    __shared__ float sA[NBB * NT * LDA];   // text tiles: 64 x 64 (padded)  ~17.0 KB
    __shared__ float sB[NPP * LDB];        // img tile:  112 x 64 (padded)  ~29.8 KB
    __shared__ float sMax[NBB * 7 * 16];   // per-(b,chunk) per-phrase maxima ~1.8 KB

    const int tid  = threadIdx.x;
    const int lane = tid & 31;
    const int wave = tid >> 5;             // 0..7 (waves 0..6 compute)
    const int row  = lane & 15;            // M for A-frag / N for B-frag
    const int koff = (lane >> 4) << 1;     // 0 or 2 (ISA 16x4 fp32 A/B layout)

    const int d  = blockIdx.x;             // img batch
    const int b0 = blockIdx.y * NBB;       // first text batch of this block

    const float* __restrict__ textB = text + (size_t)b0 * NT * NF;
    const float* __restrict__ imgD  = img  + (size_t)d  * NP * NF;

    // cooperative-load coordinates: 256 threads cover 16 rows x 64 cols (float4)
    const int lr = tid >> 4;               // 0..15
    const int lc = (tid & 15) << 2;        // 0,4,...,60

    v8f acc[NBB];
    #pragma unroll
    for (int bb = 0; bb < NBB; ++bb)
        acc[bb] = (v8f){0.f, 0.f, 0.f, 0.f, 0.f, 0.f, 0.f, 0.f};

    for (int k0 = 0; k0 < NF; k0 += KB) {
        // ---- stage 4 text tiles: 64 x 64, four float4 per thread ----
        #pragma unroll
        for (int rr = 0; rr < NBB; ++rr) {
            const float4 va =
                *(const float4*)(textB + ((size_t)rr * NT + lr) * NF + k0 + lc);
            *(float4*)(&sA[(rr * 16 + lr) * LDA + lc]) = va;
        }
        // ---- stage img tile: 112 x 64, seven float4 per thread (zero-pad i>=100) ----
        #pragma unroll
        for (int rr = 0; rr < 7; ++rr) {
            const int r = lr + rr * 16;
            float4 vb = make_float4(0.f, 0.f, 0.f, 0.f);
            if (r < NP) vb = *(const float4*)(imgD + r * NF + k0 + lc);
            *(float4*)(&sB[r * LDB + lc]) = vb;
        }
        __syncthreads();

        if (wave < 7) {   // wave-uniform branch: EXEC stays all-ones for WMMA
            const float* aBase = &sA[row * LDA + koff];
            const float* bBase = &sB[(wave * 16 + row) * LDB + koff];
            #pragma unroll
            for (int kk = 0; kk < KB; kk += 4) {
                const v2f bf = *(const v2f*)(bBase + kk);       // shared by 4 wmma
                #pragma unroll
                for (int bb = 0; bb < NBB; ++bb) {
                    const v2f af = *(const v2f*)(aBase + bb * (16 * LDA) + kk);
                    acc[bb] = __builtin_amdgcn_wmma_f32_16x16x4_f32(
                                  false, af, false, bf, (short)0, acc[bb],
                                  false, false);
                }
            }
        }
        __syncthreads();
    }

    // ---- max over proposals within this wave's 16-column chunk (per b) ----
    if (wave < 7) {
        const bool inval = (wave == 6) && ((lane & 15) >= (NP - 6 * 16)); // i >= 100
        #pragma unroll
        for (int bb = 0; bb < NBB; ++bb) {
            #pragma unroll
            for (int j = 0; j < 8; ++j) {
                float m = inval ? -__builtin_inff() : acc[bb][j];
                m = fmaxf(m, __shfl_xor(m, 1, 32));
                m = fmaxf(m, __shfl_xor(m, 2, 32));
                m = fmaxf(m, __shfl_xor(m, 4, 32));
                m = fmaxf(m, __shfl_xor(m, 8, 32)); // reduce within 16-lane half
                if ((lane & 15) == 0)               // lane 0 -> rows 0..7, lane 16 -> 8..15
                    sMax[(bb * 7 + wave) * 16 + j + ((lane >> 4) << 3)] = m;
            }
        }
    }
    __syncthreads();

    // ---- max over chunks, mean over phrases, exp: 16 lanes per output b ----
    if (tid < 16 * NBB) {
        const int bb = tid >> 4;
        const int t  = tid & 15;
        float m = -__builtin_inff();
        #pragma unroll
        for (int w = 0; w < 7; ++w) m = fmaxf(m, sMax[(bb * 7 + w) * 16 + t]);
        m += __shfl_xor(m, 1, 32);
        m += __shfl_xor(m, 2, 32);
        m += __shfl_xor(m, 4, 32);
        m += __shfl_xor(m, 8, 32);                  // sum over 16 phrases
        if (t == 0)
            out[(size_t)(b0 + bb) * NB + d] = __expf(m * (1.0f / (16.0f * 0.07f)));
    }
}

extern "C" void kernel_launch(void* const* d_in, const int* in_sizes, int n_in,
                              void* d_out, int out_size, void* d_ws, size_t ws_size,
                              hipStream_t stream) {
    const float* img  = (const float*)d_in[0];   // [128,100,512] f32
    const float* text = (const float*)d_in[1];   // [128,16,512]  f32
    // d_in[2] = labels, unused by the reference output
    float* out = (float*)d_out;                  // [128,128] f32

    dim3 grid(NB, NB / NBB);   // x = d (img batch), y = group of 4 text batches
    ContrastiveLoss_52561809768711_kernel<<<grid, 256, 0, stream>>>(img, text, out);
}